// CausalDWT_83940840833773
// MI455X (gfx1250) — compile-verified
//
#include <hip/hip_runtime.h>
#include <stdint.h>

// Problem constants (from reference): x is (32, 4096, 512) fp32.
#define B_        32
#define L_        4096
#define C_        512
#define J_        (L_ / 2)        // 2048 output pairs per (b, c)
#define P_        8               // row-pairs per block
#define ROWS_     (2 * P_)        // 16 input rows staged per tile
#define C4_       (C_ / 4)        // 128 float4 per row
#define THREADS_  256
#define CHUNKS_   ((ROWS_ * C4_) / THREADS_)   // 8 async b128 loads / thread
#define ITEMS_    ((P_ * C4_) / THREADS_)      // 4 output float4 pairs / thread
#define ROWBYTES_ (C_ * 4)                     // 2048 B per staged row
#define STRIDE_B_ (2 * ROWBYTES_)              // 4096 B per-chunk stride (2 rows)

typedef float v4f __attribute__((ext_vector_type(4)));
typedef int   i4  __attribute__((vector_size(16)));   // matches builtin param type

// ---- gfx1250 async global->LDS copy (ASYNCcnt path) -----
// CDNA5 ISA: INST_OFFSET is added to BOTH the LDS and the global
// address, so one base (global addr pair, LDS addr) + per-chunk immediate
// offset covers the whole tile with zero extra VALU.
#define ASYNC_LOAD_B128_OFF(gptr, lptr, imm)                                    \
  __builtin_amdgcn_global_load_async_to_lds_b128(                               \
      (__attribute__((address_space(1))) i4*)(gptr),                            \
      (__attribute__((address_space(3))) i4*)(lptr), (imm), 0)

#if defined(__has_builtin) && __has_builtin(__builtin_amdgcn_s_wait_asynccnt)
#define WAIT_ASYNC0() __builtin_amdgcn_s_wait_asynccnt(0)
#else
#define WAIT_ASYNC0() asm volatile("s_wait_asynccnt 0" ::: "memory")
#endif
// -----------------------------------------------------------------------------

// Recursive unroll so the per-chunk byte offset (K * STRIDE_B_) is an integer
// constant expression, as required by the builtin's immediate operand.
template <int K>
struct Loader {
  static __device__ __forceinline__ void run(const float* g, float* l, bool skip0) {
    if (!(K == 0 && skip0))                 // only K==0 has a runtime guard
      ASYNC_LOAD_B128_OFF(g, l, K * STRIDE_B_);
    Loader<K + 1>::run(g, l, skip0);
  }
};
template <>
struct Loader<CHUNKS_> {
  static __device__ __forceinline__ void run(const float*, float*, bool) {}
};

__global__ __launch_bounds__(THREADS_) void causal_dwt_kernel(
    const float* __restrict__ x, float* __restrict__ lo, float* __restrict__ hi) {
  // 16 rows x 512 floats = 32 KB staged per block.
  __shared__ __align__(16) float lds[ROWS_ * C_];

  const int tid  = threadIdx.x;
  const int blk  = blockIdx.x;
  const int b    = blk >> 8;          // blk / (J_/P_)  (J_/P_ == 256)
  const int jblk = blk & 255;
  const int j0   = jblk * P_;

  const int r0 = tid >> 7;            // this thread's first staged row (0 or 1)
  const int c4 = tid & 127;           // fixed float4 column for this thread

  // Input rows needed for pairs [j0, j0+P): t in [2*j0 - 1, 2*j0 + 2P - 2].
  const long   t0 = (long)2 * j0 - 1;
  const float* xb = x + (size_t)b * L_ * C_;

  // ---- Stage tile into LDS: 8 async b128 copies per thread ------------------
  // Rows r0+2k share one base address; chunk k adds imm offset k*4096 to both
  // the global and LDS address.
  const float* gbase = xb + (t0 + r0) * (long)C_ + (long)c4 * 4;  // k==0 row
  float*       lbase = &lds[r0 * C_ + c4 * 4];
  const bool   skip0 = (j0 == 0) && (r0 == 0);   // t == -1: padded zero row
  Loader<0>::run(gbase, lbase, skip0);

  WAIT_ASYNC0();        // each wave drains its own ASYNCcnt
  __syncthreads();      // make all waves' LDS writes visible to the block

  // ---- Compute lo/hi from LDS, store with nontemporal b128 ------------------
  const size_t out_base = ((size_t)b * J_ + j0) * C_ + (size_t)c4 * 4;
#pragma unroll
  for (int k = 0; k < ITEMS_; ++k) {
    const int jl = r0 + 2 * k;        // local pair 0..7 (this thread's share)

    // even sample: x[2j-1] -> LDS row 2*jl ; odd sample: x[2j] -> row 2*jl+1
    v4f e = {0.f, 0.f, 0.f, 0.f};
    if (!(j0 == 0 && jl == 0))
      e = *(const v4f*)&lds[(2 * jl) * C_ + c4 * 4];
    const v4f o = *(const v4f*)&lds[(2 * jl + 1) * C_ + c4 * 4];

    const v4f lo4 = e + o;
    const v4f hi4 = e - o;

    const size_t off = out_base + (size_t)jl * C_;
    __builtin_nontemporal_store(lo4, (v4f*)&lo[off]);   // streamed once, keep L2
    __builtin_nontemporal_store(hi4, (v4f*)&hi[off]);   // for the input stream
  }
}

extern "C" void kernel_launch(void* const* d_in, const int* in_sizes, int n_in,
                              void* d_out, int out_size, void* d_ws, size_t ws_size,
                              hipStream_t stream) {
  const float* x  = (const float*)d_in[0];
  float*       lo = (float*)d_out;                       // first output, flat
  float*       hi = lo + (size_t)B_ * J_ * C_;           // second output, flat

  const int grid = B_ * (J_ / P_);                       // 32 * 256 = 8192 blocks
  causal_dwt_kernel<<<grid, THREADS_, 0, stream>>>(x, lo, hi);
}